// NetEdge_41918880809703
// MI455X (gfx1250) — compile-verified
//
#include <hip/hip_runtime.h>
#include <hip/hip_bf16.h>

typedef float v2f __attribute__((ext_vector_type(2)));
typedef float v8f __attribute__((ext_vector_type(8)));

#define HSTR 68   // padded LDS row stride (floats): 4*m mod 64 distinct -> no bank conflicts

__device__ __forceinline__ v8f wmma_f32(v2f a, v2f b, v8f c) {
    // V_WMMA_F32_16X16X4_F32 : D(16x16,f32) = A(16x4,f32) x B(4x16,f32) + C
    return __builtin_amdgcn_wmma_f32_16x16x4_f32(false, a, false, b, (short)0, c, false, false);
}

// Stage a [K x 64] row-major weight into LDS in B-fragment order:
// dst[(kk*4 + j)*64 + lh*32 + ll*2 + p] = src[(kk*4 + 2*lh + p)*64 + j*16 + ll]
// so each WMMA B fragment is one aligned ds_load_b64, wave reads 256B contiguous.
__device__ __forceinline__ void stage_w_frag(const float* __restrict__ src, float* dst,
                                             int K, int tid, int nthreads) {
    for (int i = tid; i < K * 64; i += nthreads) {
        int k = i >> 6, n = i & 63;
        int kk = k >> 2, r = k & 3;
        int d = ((kk * 4 + (n >> 4)) << 6) + ((r >> 1) << 5) + ((n & 15) << 1) + (r & 1);
        dst[d] = src[i];
    }
}

// ---------------------------------------------------------------- edge kernels

// xc[n][0:64] = x[n][:], xc[n][64:96] = 0
__global__ void build_xc_kernel(const float* __restrict__ x, float* __restrict__ xc, int n) {
    int id = blockIdx.x * blockDim.x + threadIdx.x;
    if (id >= n * 96) return;
    int node = id / 96, c = id - node * 96;
    xc[id] = (c < 64) ? x[(size_t)node * 64 + c] : 0.0f;
}

// xc[row[e]][64+c] += edge_attr[e][c]   (fused edge segment_sum + concat)
__global__ void edge_aggr_kernel(const float* __restrict__ edge_attr,
                                 const int* __restrict__ row,
                                 float* __restrict__ xc, int ne) {
    int id = blockIdx.x * blockDim.x + threadIdx.x;
    if (id >= ne * 32) return;
    int e = id >> 5, c = id & 31;
    atomicAdd(&xc[(size_t)row[e] * 96 + 64 + c], edge_attr[(size_t)e * 32 + c]);
}

// agg[col[e]][c] += xc[row[e]][c], c in [0,96)
__global__ void scatter_kernel(const float* __restrict__ xc,
                               const int* __restrict__ row,
                               const int* __restrict__ col,
                               float* __restrict__ agg, int ne) {
    int id = blockIdx.x * blockDim.x + threadIdx.x;
    if (id >= ne * 96) return;
    int e = id / 96, c = id - e * 96;
    atomicAdd(&agg[(size_t)col[e] * 96 + c], xc[(size_t)row[e] * 96 + c]);
}

// pooled[batch[n]][c] += xnode[n][c]
__global__ void pool_kernel(const float* __restrict__ xnode,
                            const int* __restrict__ batch,
                            float* __restrict__ pooled, int n) {
    int id = blockIdx.x * blockDim.x + threadIdx.x;
    if (id >= n * 64) return;
    int node = id >> 6, c = id & 63;
    atomicAdd(&pooled[(size_t)batch[node] * 64 + c], xnode[id]);
}

// ---------------------------------------------------------------- layer MLP (WMMA)
// xout = relu( mlp(agg, rel) + mlp(xc, root) ), mlp(h) = relu(h@W1+b1)@W2+b2
// One wave per 16-row node tile; d = 96, H = 64.
__global__ void mlp_layer_kernel(const float* __restrict__ agg,
                                 const float* __restrict__ xc,
                                 const float* __restrict__ w1rel_g, const float* __restrict__ b1rel_g,
                                 const float* __restrict__ w2rel_g, const float* __restrict__ b2rel_g,
                                 const float* __restrict__ w1root_g, const float* __restrict__ b1root_g,
                                 const float* __restrict__ w2root_g, const float* __restrict__ b2root_g,
                                 float* __restrict__ xout, int ntiles) {
    extern __shared__ float lds[];
    float* w1rel  = lds;                    // 96*64  (fragment order)
    float* w1root = w1rel  + 96 * 64;       // 96*64
    float* w2rel  = w1root + 96 * 64;       // 64*64
    float* w2root = w2rel  + 64 * 64;       // 64*64
    float* b1rel  = w2root + 64 * 64;       // 64
    float* b1root = b1rel + 64;             // 64
    float* b2sum  = b1root + 64;            // 64
    float* hbase  = b2sum + 64;             // 8 waves * 16*HSTR

    int tid = threadIdx.x;
    stage_w_frag(w1rel_g,  w1rel,  96, tid, blockDim.x);
    stage_w_frag(w1root_g, w1root, 96, tid, blockDim.x);
    stage_w_frag(w2rel_g,  w2rel,  64, tid, blockDim.x);
    stage_w_frag(w2root_g, w2root, 64, tid, blockDim.x);
    if (tid < 64) { b1rel[tid] = b1rel_g[tid]; b1root[tid] = b1root_g[tid];
                    b2sum[tid] = b2rel_g[tid] + b2root_g[tid]; }
    __syncthreads();

    int wave = tid >> 5, lane = tid & 31;
    int ll = lane & 15, lh = lane >> 4;
    int boff = lh * 32 + (ll << 1);           // lane offset into each 64-float B fragment
    int tile = blockIdx.x * (blockDim.x >> 5) + wave;
    if (tile >= ntiles) return;               // whole wave exits -> EXEC all-ones elsewhere
    int r0 = tile * 16;
    float* hbuf = hbase + wave * (16 * HSTR);

    v8f outc[4] = {};                         // accumulates both paths' second GEMM
    #pragma unroll
    for (int p = 0; p < 2; ++p) {
        const float* X  = p ? xc     : agg;
        const float* W1 = p ? w1root : w1rel;
        const float* B1 = p ? b1root : b1rel;
        const float* W2 = p ? w2root : w2rel;

        v8f c[4] = {};
        // GEMM1: [16 x 96] @ [96 x 64]
        for (int kk = 0; kk < 24; ++kk) {
            int k0 = kk * 4 + lh * 2;
            v2f a = *(const v2f*)(X + (size_t)(r0 + ll) * 96 + k0);
            #pragma unroll
            for (int j = 0; j < 4; ++j) {
                v2f b = *(const v2f*)(W1 + ((kk * 4 + j) << 6) + boff);
                c[j] = wmma_f32(a, b, c[j]);
            }
        }
        // bias + ReLU -> LDS hidden tile (C-layout -> row-major, padded stride)
        #pragma unroll
        for (int j = 0; j < 4; ++j) {
            float bb = B1[j * 16 + ll];
            #pragma unroll
            for (int v = 0; v < 8; ++v) {
                float hv = c[j][v] + bb;
                hbuf[(v + 8 * lh) * HSTR + j * 16 + ll] = hv > 0.0f ? hv : 0.0f;
            }
        }
        // GEMM2: [16 x 64] @ [64 x 64], A re-read in A-layout from LDS
        for (int kk = 0; kk < 16; ++kk) {
            int k0 = kk * 4 + lh * 2;
            v2f a = *(const v2f*)(hbuf + ll * HSTR + k0);
            #pragma unroll
            for (int j = 0; j < 4; ++j) {
                v2f b = *(const v2f*)(W2 + ((kk * 4 + j) << 6) + boff);
                outc[j] = wmma_f32(a, b, outc[j]);
            }
        }
    }
    // combined bias + outer ReLU + store
    #pragma unroll
    for (int j = 0; j < 4; ++j) {
        float bb = b2sum[j * 16 + ll];
        #pragma unroll
        for (int v = 0; v < 8; ++v) {
            float o = outc[j][v] + bb;
            xout[(size_t)(r0 + v + 8 * lh) * 64 + j * 16 + ll] = o > 0.0f ? o : 0.0f;
        }
    }
}

// ---------------------------------------------------------------- final MLP (WMMA)
// out[G,10] = relu(pooled@W1+b1) @ W2 + b2
__global__ void mlp_final_kernel(const float* __restrict__ pooled,
                                 const float* __restrict__ w1_g, const float* __restrict__ b1_g,
                                 const float* __restrict__ w2_g, const float* __restrict__ b2_g,
                                 float* __restrict__ out, int g) {
    extern __shared__ float lds[];
    float* w1 = lds;            // 64*64 (fragment order)
    float* w2 = w1 + 64 * 64;   // 16*64 fragment order, zero-padded cols 10..15
    float* b1 = w2 + 16 * 64;   // 64
    float* b2 = b1 + 64;        // 16 (padded)
    float* hbase = b2 + 16;     // 8 waves * 16*HSTR

    int tid = threadIdx.x;
    stage_w_frag(w1_g, w1, 64, tid, blockDim.x);
    // W2 is [64 x 10]; stage into fragment order padded to 16 columns with zeros.
    for (int i = tid; i < 16 * 64; i += blockDim.x) {
        int kk = i >> 6, rem = i & 63;
        int lh = rem >> 5, q = rem & 31, ll = q >> 1, p = q & 1;
        int k = kk * 4 + 2 * lh + p;
        w2[i] = (ll < 10) ? w2_g[k * 10 + ll] : 0.0f;
    }
    if (tid < 64) b1[tid] = b1_g[tid];
    if (tid < 16) b2[tid] = (tid < 10) ? b2_g[tid] : 0.0f;
    __syncthreads();

    int wave = tid >> 5, lane = tid & 31;
    int ll = lane & 15, lh = lane >> 4;
    int boff = lh * 32 + (ll << 1);
    int tile = blockIdx.x * (blockDim.x >> 5) + wave;
    if (tile * 16 >= g) return;
    int r0 = tile * 16;
    float* hbuf = hbase + wave * (16 * HSTR);

    // GEMM1: [16 x 64] @ [64 x 64]
    v8f c[4] = {};
    for (int kk = 0; kk < 16; ++kk) {
        int k0 = kk * 4 + lh * 2;
        v2f a = *(const v2f*)(pooled + (size_t)(r0 + ll) * 64 + k0);
        #pragma unroll
        for (int j = 0; j < 4; ++j) {
            v2f b = *(const v2f*)(w1 + ((kk * 4 + j) << 6) + boff);
            c[j] = wmma_f32(a, b, c[j]);
        }
    }
    #pragma unroll
    for (int j = 0; j < 4; ++j) {
        float bb = b1[j * 16 + ll];
        #pragma unroll
        for (int v = 0; v < 8; ++v) {
            float hv = c[j][v] + bb;
            hbuf[(v + 8 * lh) * HSTR + j * 16 + ll] = hv > 0.0f ? hv : 0.0f;
        }
    }
    // GEMM2: [16 x 64] @ [64 x 10]  (single N-tile, zero-padded columns)
    v8f o = {};
    for (int kk = 0; kk < 16; ++kk) {
        int k0 = kk * 4 + lh * 2;
        v2f a = *(const v2f*)(hbuf + ll * HSTR + k0);
        v2f b = *(const v2f*)(w2 + (kk << 6) + boff);
        o = wmma_f32(a, b, o);
    }
    if (ll < 10) {
        float bb = b2[ll];
        #pragma unroll
        for (int v = 0; v < 8; ++v)
            out[(size_t)(r0 + v + 8 * lh) * 10 + ll] = o[v] + bb;
    }
}

// ---------------------------------------------------------------- host

extern "C" void kernel_launch(void* const* d_in, const int* in_sizes, int n_in,
                              void* d_out, int out_size, void* d_ws, size_t ws_size,
                              hipStream_t stream) {
    const float* x_in      = (const float*)d_in[0];
    const float* edge_attr = (const float*)d_in[1];
    const float* relW1_0 = (const float*)d_in[2],  *relb1_0 = (const float*)d_in[3];
    const float* relW2_0 = (const float*)d_in[4],  *relb2_0 = (const float*)d_in[5];
    const float* rooW1_0 = (const float*)d_in[6],  *roob1_0 = (const float*)d_in[7];
    const float* rooW2_0 = (const float*)d_in[8],  *roob2_0 = (const float*)d_in[9];
    const float* relW1_1 = (const float*)d_in[10], *relb1_1 = (const float*)d_in[11];
    const float* relW2_1 = (const float*)d_in[12], *relb2_1 = (const float*)d_in[13];
    const float* rooW1_1 = (const float*)d_in[14], *roob1_1 = (const float*)d_in[15];
    const float* rooW2_1 = (const float*)d_in[16], *roob2_1 = (const float*)d_in[17];
    const float* finW1 = (const float*)d_in[18], *finb1 = (const float*)d_in[19];
    const float* finW2 = (const float*)d_in[20], *finb2 = (const float*)d_in[21];
    const int* eidx  = (const int*)d_in[22];
    const int* batch = (const int*)d_in[23];

    const int N  = in_sizes[0] / 64;
    const int E  = in_sizes[1] / 32;
    const int G  = out_size / 10;
    const int* row = eidx;
    const int* col = eidx + E;

    float* xc     = (float*)d_ws;          // N*96
    float* agg    = xc  + (size_t)N * 96;  // N*96
    float* xbuf   = agg + (size_t)N * 96;  // N*64
    float* pooled = xbuf + (size_t)N * 64; // G*64

    const int ntiles = (N + 15) / 16;
    const int mlpBlocks = (ntiles + 7) / 8;
    const size_t layerLds = (size_t)(96*64*2 + 64*64*2 + 64*3 + 8*16*HSTR) * sizeof(float);
    const size_t finalLds = (size_t)(64*64 + 16*64 + 64 + 16 + 8*16*HSTR) * sizeof(float);

    for (int l = 0; l < 2; ++l) {
        const float* xl = (l == 0) ? x_in : xbuf;
        build_xc_kernel<<<(N * 96 + 255) / 256, 256, 0, stream>>>(xl, xc, N);
        edge_aggr_kernel<<<(E * 32 + 255) / 256, 256, 0, stream>>>(edge_attr, row, xc, E);
        hipMemsetAsync(agg, 0, (size_t)N * 96 * sizeof(float), stream);
        scatter_kernel<<<(E * 96 + 255) / 256, 256, 0, stream>>>(xc, row, col, agg, E);
        if (l == 0)
            mlp_layer_kernel<<<mlpBlocks, 256, layerLds, stream>>>(
                agg, xc, relW1_0, relb1_0, relW2_0, relb2_0,
                rooW1_0, roob1_0, rooW2_0, roob2_0, xbuf, ntiles);
        else
            mlp_layer_kernel<<<mlpBlocks, 256, layerLds, stream>>>(
                agg, xc, relW1_1, relb1_1, relW2_1, relb2_1,
                rooW1_1, roob1_1, rooW2_1, roob2_1, xbuf, ntiles);
    }
    hipMemsetAsync(pooled, 0, (size_t)G * 64 * sizeof(float), stream);
    pool_kernel<<<(N * 64 + 255) / 256, 256, 0, stream>>>(xbuf, batch, pooled, N);
    mlp_final_kernel<<<(G / 16 + 7) / 8, 256, finalLds, stream>>>(
        pooled, finW1, finb1, finW2, finb2, (float*)d_out, G);
}